// SpatialRNN_7687991460176
// MI455X (gfx1250) — compile-verified
//
#include <hip/hip_runtime.h>
#include <stdint.h>

// SpatialRNN along W: h_w = relu(alpha[c]*h_{w-1} + x[n,c,h,w]), out[n,c,h,w] = h_w.
// N=16, C=256, H=96, W=96, float32. Memory-bound: ~302 MB @ 23.3 TB/s ~ 13 us.
// Strategy: async global->LDS tile copy (coalesced B128, ASYNCcnt), sequential
// ReLU scan in a padded LDS tile (vectorizes to ds_load_b128/ds_store_b128),
// async LDS->global store. Each wave32 owns 32 contiguous rows -> no barriers.

#define LDS_AS __attribute__((address_space(3)))
#define GLB_AS __attribute__((address_space(1)))

typedef int v4i __attribute__((ext_vector_type(4)));
typedef GLB_AS v4i* gv4i_ptr;
typedef LDS_AS v4i* lv4i_ptr;

constexpr int Wdim = 96;
constexpr int Hdim = 96;
constexpr int Cdim = 256;
constexpr int Ndim = 16;
constexpr int ROWS_PER_BLOCK = 128;          // == blockDim.x, 4 waves
constexpr int PITCH = 100;                   // floats; rows 16B-aligned, low bank conflict
constexpr int CHUNKS_PER_ROW = Wdim / 4;     // 24 float4 chunks per row
constexpr int CHUNKS_PER_LANE = CHUNKS_PER_ROW; // (32 rows * 24 chunks) / 32 lanes

__device__ __forceinline__ void async_load_b128(float* lds, const float* gsrc) {
#if __has_builtin(__builtin_amdgcn_global_load_async_to_lds_b128)
  const GLB_AS float* g1 = (const GLB_AS float*)gsrc;  // addrspacecast
  LDS_AS float*       l1 = (LDS_AS float*)lds;         // addrspacecast
  __builtin_amdgcn_global_load_async_to_lds_b128(
      (gv4i_ptr)g1, (lv4i_ptr)l1, 0, 0);
#else
  unsigned loff = (unsigned)(uintptr_t)(LDS_AS float*)lds;
  asm volatile("global_load_async_to_lds_b128 %0, %1, off"
               :: "v"(loff), "v"((unsigned long long)(uintptr_t)gsrc)
               : "memory");
#endif
}

__device__ __forceinline__ void async_store_b128(float* gdst, const float* lds) {
#if __has_builtin(__builtin_amdgcn_global_store_async_from_lds_b128)
  GLB_AS float*       g1 = (GLB_AS float*)gdst;        // addrspacecast
  const LDS_AS float* l1 = (const LDS_AS float*)lds;   // addrspacecast
  __builtin_amdgcn_global_store_async_from_lds_b128(
      (gv4i_ptr)g1, (lv4i_ptr)l1, 0, 0);
#else
  unsigned loff = (unsigned)(uintptr_t)(const LDS_AS float*)lds;
  asm volatile("global_store_async_from_lds_b128 %1, %0, off"
               :: "v"(loff), "v"((unsigned long long)(uintptr_t)gdst)
               : "memory");
#endif
}

__device__ __forceinline__ void wait_asynccnt0() {
#if __has_builtin(__builtin_amdgcn_s_wait_asynccnt)
  __builtin_amdgcn_s_wait_asynccnt(0);
  asm volatile("" ::: "memory");   // compiler fence: keep LDS reads below the HW wait
#else
  asm volatile("s_wait_asynccnt 0" ::: "memory");
#endif
}

__device__ __forceinline__ void wait_dscnt0() {
  asm volatile("s_wait_dscnt 0" ::: "memory");
}

__global__ __launch_bounds__(ROWS_PER_BLOCK)
void spatial_rnn_scan_kernel(const float* __restrict__ x,
                             const float* __restrict__ alpha,
                             float* __restrict__ out) {
  __shared__ float tile[ROWS_PER_BLOCK * PITCH];   // 51200 B

  const int tid  = threadIdx.x;
  const int lane = tid & 31;
  const int wv   = tid >> 5;
  const int waveRow0 = wv * 32;                    // row within block
  const long long blockRow0 = (long long)blockIdx.x * ROWS_PER_BLOCK;

  const float* gsrc = x   + (blockRow0 + waveRow0) * Wdim;  // wave's 32*96 floats (contiguous)
  float*       gdst = out + (blockRow0 + waveRow0) * Wdim;

  // ---- Phase 1: async copy 32 rows x 96 floats into padded LDS tile ----
  #pragma unroll
  for (int i = 0; i < CHUNKS_PER_LANE; ++i) {
    const int chunk = i * 32 + lane;               // 0..767
    const int r  = chunk / CHUNKS_PER_ROW;         // 0..31
    const int c4 = chunk - r * CHUNKS_PER_ROW;     // 0..23
    async_load_b128(&tile[(waveRow0 + r) * PITCH + c4 * 4], gsrc + chunk * 4);
  }
  wait_asynccnt0();

  // ---- Phase 2: sequential ReLU scan along W, one row per lane, in LDS ----
  const int row = waveRow0 + lane;
  const long long grow = blockRow0 + row;          // global flat row id
  const float a = alpha[(int)((grow / Hdim) % Cdim)];  // wave-uniform
  float* rp = &tile[row * PITCH];
  float h = 0.0f;
  #pragma unroll 4
  for (int w = 0; w < Wdim; ++w) {
    h = fmaxf(fmaf(a, h, rp[w]), 0.0f);
    rp[w] = h;
  }
  wait_dscnt0();  // scan's LDS writes complete before the async store engine reads

  // ---- Phase 3: async store LDS tile -> out (coalesced B128) ----
  #pragma unroll
  for (int i = 0; i < CHUNKS_PER_LANE; ++i) {
    const int chunk = i * 32 + lane;
    const int r  = chunk / CHUNKS_PER_ROW;
    const int c4 = chunk - r * CHUNKS_PER_ROW;
    async_store_b128(gdst + chunk * 4, &tile[(waveRow0 + r) * PITCH + c4 * 4]);
  }
  wait_asynccnt0();  // stores complete before wave retires (S_ENDPGM also wait-idles)
}

extern "C" void kernel_launch(void* const* d_in, const int* in_sizes, int n_in,
                              void* d_out, int out_size, void* d_ws, size_t ws_size,
                              hipStream_t stream) {
  (void)in_sizes; (void)n_in; (void)d_ws; (void)ws_size; (void)out_size;
  const float* x     = (const float*)d_in[0];   // (16,256,96,96) f32
  const float* alpha = (const float*)d_in[1];   // (256,) f32
  float* out = (float*)d_out;

  const int totalRows = Ndim * Cdim * Hdim;      // 393216
  const int blocks = totalRows / ROWS_PER_BLOCK; // 3072
  spatial_rnn_scan_kernel<<<blocks, ROWS_PER_BLOCK, 0, stream>>>(x, alpha, out);
}